// RNN_ILSTM_62929860821092
// MI455X (gfx1250) — compile-verified
//
#include <hip/hip_runtime.h>
#include <math.h>

// ---- WMMA types (CDNA5, wave32) ----
typedef __attribute__((ext_vector_type(2))) float v2f;
typedef __attribute__((ext_vector_type(8))) float v8f;

#define WMMA_F32X4(A, B, C) \
    __builtin_amdgcn_wmma_f32_16x16x4_f32(false, (A), false, (B), (short)0, (C), false, false)

// Problem dims
#define TT 64
#define BB 64
#define DD 512
#define HH 512
#define OO 128
#define G4H 2048   // 4*H

__device__ __forceinline__ float sigmoidf_(float x) {
    return 1.0f / (1.0f + __expf(-x));
}

// ---------------------------------------------------------------------------
// Kernel 0: reset grid-barrier counter (ws is not re-poisoned between replays)
// ---------------------------------------------------------------------------
__global__ void init_counter_kernel(unsigned* counter) {
    if (threadIdx.x == 0) *counter = 0u;
}

// ---------------------------------------------------------------------------
// Kernel 1: gates_x[t,b,j] = sum_d x[t,b,d]*w_ih[t,j,d] + b_ih[t,j] + b_hh[t,j]
// One wave per (t, 16-wide j tile): M=64 (4 sub-tiles) x N=16, K=512.
// Streams w_ih (256 MB) exactly once, non-temporally (read-once data);
// x (8 MB) stays L2-resident and is reused by all 128 j-tiles per t.
// ---------------------------------------------------------------------------
__global__ void __launch_bounds__(32)
gates_x_kernel(const float* __restrict__ x,
               const float* __restrict__ w_ih,
               const float* __restrict__ b_ih,
               const float* __restrict__ b_hh,
               float* __restrict__ gx) {
    const int n    = blockIdx.x;          // 0..127 : 16-col tile over 4H
    const int t    = blockIdx.y;          // 0..63
    const int lane = threadIdx.x & 31;
    const int l16  = lane & 15;
    const int half = lane >> 4;
    const int j    = n * 16 + l16;        // output column == row of w_ih[t]

    const float* W = w_ih + ((size_t)t * G4H + j) * DD + half * 2;
    const float* X = x    + ((size_t)t * BB) * DD + half * 2;

    v8f acc0 = {}, acc1 = {}, acc2 = {}, acc3 = {};

#pragma unroll 4
    for (int k = 0; k < DD; k += 4) {
        v2f bm = __builtin_nontemporal_load((const v2f*)(W + k));   // stream NT
        v2f a0 = *(const v2f*)(X + (size_t)(l16     ) * DD + k);
        v2f a1 = *(const v2f*)(X + (size_t)(l16 + 16) * DD + k);
        v2f a2 = *(const v2f*)(X + (size_t)(l16 + 32) * DD + k);
        v2f a3 = *(const v2f*)(X + (size_t)(l16 + 48) * DD + k);
        acc0 = WMMA_F32X4(a0, bm, acc0);
        acc1 = WMMA_F32X4(a1, bm, acc1);
        acc2 = WMMA_F32X4(a2, bm, acc2);
        acc3 = WMMA_F32X4(a3, bm, acc3);
    }

    const float bias = b_ih[t * G4H + j] + b_hh[t * G4H + j];
    float* Gp = gx + ((size_t)t * BB) * G4H + j;
#pragma unroll
    for (int r = 0; r < 8; ++r) {
        const int m0 = half * 8 + r;            // C/D layout: row = 8*half + vgpr
        Gp[(size_t)(m0     ) * G4H] = acc0[r] + bias;
        Gp[(size_t)(m0 + 16) * G4H] = acc1[r] + bias;
        Gp[(size_t)(m0 + 32) * G4H] = acc2[r] + bias;
        Gp[(size_t)(m0 + 48) * G4H] = acc3[r] + bias;
    }
}

// ---------------------------------------------------------------------------
// Grid-wide barrier (persistent kernel, 32 co-resident blocks)
// ---------------------------------------------------------------------------
__device__ __forceinline__ void grid_barrier(unsigned* counter, unsigned target) {
    __threadfence();
    __syncthreads();
    if (threadIdx.x == 0) {
        atomicAdd(counter, 1u);
        while (__hip_atomic_load(counter, __ATOMIC_RELAXED, __HIP_MEMORY_SCOPE_AGENT) < target) {
            __builtin_amdgcn_s_sleep(1);
        }
    }
    __syncthreads();
    __threadfence();
}

// ---------------------------------------------------------------------------
// Kernel 2: persistent recurrence. 32 blocks x 512 threads (16 waves).
// Block n owns h-columns [16n, 16n+16). Wave (kq,g) computes the K-quarter
// kq of gate g's 16-wide tile of h_{t-1} @ w_hh[t].T  (M=64 x N=16 x K=128)
// via f32 WMMA; partials reduced through LDS. Cell state c lives in LDS and
// never touches HBM. h double-buffered in global; one grid barrier per step.
// 512 waves in flight streaming w_hh (4 MB/step) to track the HBM floor.
// ---------------------------------------------------------------------------
__global__ void __launch_bounds__(512)
lstm_recurrent_kernel(const float* __restrict__ gx,
                      const float* __restrict__ w_hh,
                      const int*   __restrict__ lengths,
                      float* __restrict__ hA,
                      float* __restrict__ hB,
                      float* __restrict__ hlast,
                      unsigned* __restrict__ counter) {
    __shared__ float gl[4][4][BB][16];  // [kq][gate][b][c] partial sums (64 KB)
    __shared__ float cl[BB][16];        // cell state, this block's cols  (4 KB)
    __shared__ int   llen[BB];

    const int nblk = gridDim.x;             // 32
    const int n    = blockIdx.x;            // column tile over H
    const int wv   = threadIdx.x >> 5;      // 0..15
    const int g    = wv & 3;                // gate id (i,f,g,o)
    const int kq   = wv >> 2;               // K quarter
    const int lane = threadIdx.x & 31;
    const int l16  = lane & 15;
    const int half = lane >> 4;

    // init c (LDS), lengths (LDS) and this block's columns of both h buffers
    for (int idx = threadIdx.x; idx < BB * 16; idx += 512) {
        const int b = idx >> 4, c = idx & 15;
        cl[b][c] = 0.0f;
        hA[(size_t)b * HH + n * 16 + c] = 0.0f;
        hB[(size_t)b * HH + n * 16 + c] = 0.0f;
    }
    if (threadIdx.x < BB) llen[threadIdx.x] = lengths[threadIdx.x];
    grid_barrier(counter, (unsigned)nblk);   // h fully zeroed everywhere

    const float* hcur = hA;
    float*       hnxt = hB;

    const int j  = g * HH + n * 16 + l16;    // row of w_hh[t]
    const int k0 = kq * 128;                 // this wave's K range

    for (int t = 0; t < TT; ++t) {
        const float* W  = w_hh + ((size_t)t * G4H + j) * HH + k0 + half * 2;
        const float* Hc = hcur + k0 + half * 2;

        v8f acc0 = {}, acc1 = {}, acc2 = {}, acc3 = {};
#pragma unroll 4
        for (int k = 0; k < 128; k += 4) {
            v2f bm = __builtin_nontemporal_load((const v2f*)(W + k));  // stream NT
            v2f a0 = *(const v2f*)(Hc + (size_t)(l16     ) * HH + k);
            v2f a1 = *(const v2f*)(Hc + (size_t)(l16 + 16) * HH + k);
            v2f a2 = *(const v2f*)(Hc + (size_t)(l16 + 32) * HH + k);
            v2f a3 = *(const v2f*)(Hc + (size_t)(l16 + 48) * HH + k);
            acc0 = WMMA_F32X4(a0, bm, acc0);
            acc1 = WMMA_F32X4(a1, bm, acc1);
            acc2 = WMMA_F32X4(a2, bm, acc2);
            acc3 = WMMA_F32X4(a3, bm, acc3);
        }

        // park K-partial gate tiles in LDS
#pragma unroll
        for (int r = 0; r < 8; ++r) {
            const int m0 = half * 8 + r;
            gl[kq][g][m0     ][l16] = acc0[r];
            gl[kq][g][m0 + 16][l16] = acc1[r];
            gl[kq][g][m0 + 32][l16] = acc2[r];
            gl[kq][g][m0 + 48][l16] = acc3[r];
        }
        __syncthreads();

        // LSTM cell update for this block's 64x16 slab (reduce 4 K-partials,
        // add precomputed x-gates+biases from gx)
        const float* Gt = gx + ((size_t)t * BB) * G4H;
        for (int idx = threadIdx.x; idx < BB * 16; idx += 512) {
            const int b = idx >> 4, c = idx & 15;
            const int col = n * 16 + c;
            const float gi = Gt[(size_t)b * G4H + 0 * HH + col]
                           + gl[0][0][b][c] + gl[1][0][b][c]
                           + gl[2][0][b][c] + gl[3][0][b][c];
            const float gf = Gt[(size_t)b * G4H + 1 * HH + col]
                           + gl[0][1][b][c] + gl[1][1][b][c]
                           + gl[2][1][b][c] + gl[3][1][b][c];
            const float gg = Gt[(size_t)b * G4H + 2 * HH + col]
                           + gl[0][2][b][c] + gl[1][2][b][c]
                           + gl[2][2][b][c] + gl[3][2][b][c];
            const float go = Gt[(size_t)b * G4H + 3 * HH + col]
                           + gl[0][3][b][c] + gl[1][3][b][c]
                           + gl[2][3][b][c] + gl[3][3][b][c];

            const float ig = sigmoidf_(gi);
            const float fg = sigmoidf_(gf);
            const float gv = tanhf(gg);
            const float og = sigmoidf_(go);
            const float cn = fg * cl[b][c] + ig * gv;
            cl[b][c] = cn;
            const float h = og * tanhf(cn);
            hnxt[(size_t)b * HH + col] = h;
            if (llen[b] - 1 == t)
                hlast[(size_t)b * HH + col] = h;
        }
        __syncthreads();                                   // gl/cl reuse
        grid_barrier(counter, (unsigned)(nblk * (t + 2))); // h_t visible everywhere

        float* tmp = (float*)hcur; hcur = hnxt; hnxt = tmp;
    }
}

// ---------------------------------------------------------------------------
// Kernel 3: logits = h_last @ w_ho.T + b_ho ; out = log_softmax(logits)
// One block, 8 waves; wave w -> 16 output classes via f32 WMMA.
// ---------------------------------------------------------------------------
__global__ void __launch_bounds__(256)
classifier_kernel(const float* __restrict__ hlast,
                  const float* __restrict__ w_ho,
                  const float* __restrict__ b_ho,
                  float* __restrict__ out) {
    __shared__ float logits[BB][OO];   // 32 KB

    const int w    = threadIdx.x >> 5;   // 0..7 : 16-class tile
    const int lane = threadIdx.x & 31;
    const int l16  = lane & 15;
    const int half = lane >> 4;
    const int j    = w * 16 + l16;       // output class

    const float* W  = w_ho + (size_t)j * HH + half * 2;
    const float* Hc = hlast + half * 2;

    v8f acc0 = {}, acc1 = {}, acc2 = {}, acc3 = {};
#pragma unroll 4
    for (int k = 0; k < HH; k += 4) {
        v2f bm = *(const v2f*)(W + k);
        v2f a0 = *(const v2f*)(Hc + (size_t)(l16     ) * HH + k);
        v2f a1 = *(const v2f*)(Hc + (size_t)(l16 + 16) * HH + k);
        v2f a2 = *(const v2f*)(Hc + (size_t)(l16 + 32) * HH + k);
        v2f a3 = *(const v2f*)(Hc + (size_t)(l16 + 48) * HH + k);
        acc0 = WMMA_F32X4(a0, bm, acc0);
        acc1 = WMMA_F32X4(a1, bm, acc1);
        acc2 = WMMA_F32X4(a2, bm, acc2);
        acc3 = WMMA_F32X4(a3, bm, acc3);
    }

    const float bias = b_ho[j];
#pragma unroll
    for (int r = 0; r < 8; ++r) {
        const int m0 = half * 8 + r;
        logits[m0     ][j] = acc0[r] + bias;
        logits[m0 + 16][j] = acc1[r] + bias;
        logits[m0 + 32][j] = acc2[r] + bias;
        logits[m0 + 48][j] = acc3[r] + bias;
    }
    __syncthreads();

    if (threadIdx.x < BB) {
        const int b = threadIdx.x;
        float mx = -INFINITY;
        for (int o = 0; o < OO; ++o) mx = fmaxf(mx, logits[b][o]);
        float s = 0.0f;
        for (int o = 0; o < OO; ++o) s += __expf(logits[b][o] - mx);
        const float ls = __logf(s);
        for (int o = 0; o < OO; ++o)
            out[(size_t)b * OO + o] = logits[b][o] - mx - ls;
    }
}

// ---------------------------------------------------------------------------
// Launch
// ---------------------------------------------------------------------------
extern "C" void kernel_launch(void* const* d_in, const int* in_sizes, int n_in,
                              void* d_out, int out_size, void* d_ws, size_t ws_size,
                              hipStream_t stream) {
    const float* x      = (const float*)d_in[0];
    const float* w_ih   = (const float*)d_in[1];
    const float* w_hh   = (const float*)d_in[2];
    const float* b_ih   = (const float*)d_in[3];
    const float* b_hh   = (const float*)d_in[4];
    const float* w_ho   = (const float*)d_in[5];
    const float* b_ho   = (const float*)d_in[6];
    const int*   lens   = (const int*)d_in[7];
    float*       out    = (float*)d_out;

    // workspace layout
    float* gx    = (float*)d_ws;                       // T*B*4H   = 32 MB
    float* hA    = gx + (size_t)TT * BB * G4H;         // B*H
    float* hB    = hA + (size_t)BB * HH;               // B*H
    float* hlast = hB + (size_t)BB * HH;               // B*H
    unsigned* counter = (unsigned*)(hlast + (size_t)BB * HH);

    init_counter_kernel<<<1, 32, 0, stream>>>(counter);

    gates_x_kernel<<<dim3(G4H / 16, TT), 32, 0, stream>>>(x, w_ih, b_ih, b_hh, gx);

    lstm_recurrent_kernel<<<HH / 16, 512, 0, stream>>>(gx, w_hh, lens,
                                                       hA, hB, hlast, counter);

    classifier_kernel<<<1, 256, 0, stream>>>(hlast, w_ho, b_ho, out);
}